// GAT_37203006718142
// MI455X (gfx1250) — compile-verified
//
#include <hip/hip_runtime.h>

// ---------------------------------------------------------------------------
// GAT 2-layer forward for MI455X (gfx1250, wave32, WMMA).
// N=50000 (=3125*16), E=800000, F_IN=128, F_OUT=32, H=4, H*F_OUT=128.
// ---------------------------------------------------------------------------

typedef float v2f __attribute__((ext_vector_type(2)));
typedef float v4f __attribute__((ext_vector_type(4)));
typedef float v8f __attribute__((ext_vector_type(8)));

#define NEG_SLOPE 0.2f
#define LN_EPS 1e-5f

// ---- monotone float<->uint encoding for atomic segment-max ----------------
__device__ __forceinline__ unsigned fenc(float f) {
  unsigned u = __float_as_uint(f);
  return (u >> 31) ? ~u : (u | 0x80000000u);
}
__device__ __forceinline__ float fdec(unsigned u) {
  return __uint_as_float((u >> 31) ? (u & 0x7FFFFFFFu) : ~u);
}
__device__ __forceinline__ float lrelu(float x) {
  return x > 0.f ? x : x * NEG_SLOPE;
}

// ---------------------------------------------------------------------------
// GEMM:  C[M x 128] = A[M x K] * B[K x 128]   via V_WMMA_F32_16X16X4_F32.
// One wave per 16-row stripe, 8 col-tiles (full 128 cols), K multiple of 4.
//
// B is staged once per block into LDS, pre-swizzled so that per lane the
// eight col-tile fragments of one k-step are 16 contiguous floats:
//     bs[ks][lane][c][2]
// -> four ds_load_b128 per k-step (issuable in one clause, one dscnt wait
//    before the 8-WMMA burst). v4f is 16B aligned => even VGPR, so .xy/.zw
//    are valid even-aligned v2f WMMA operands.
//
// Fragment layouts (ISA 7.12.2):
//   A (16x4 f32): lanes 0-15 -> {K+0,K+1}, lanes 16-31 -> {K+2,K+3}, m=lane&15
//   B (4x16 f32): same K split, n = lane&15
//   D (16x16 f32): vgpr r -> row r + 8*(lane>=16), col = lane&15
// ---------------------------------------------------------------------------
template <int K>
__global__ void gat_gemm_wmma(const float* __restrict__ Amat,
                              const float* __restrict__ Bmat,
                              float* __restrict__ Cmat,
                              int M) {
  __shared__ float bs[K * 128];  // (K/4) ksteps * 32 lanes * 8 ctiles * 2

  const int tid = threadIdx.x;

  // swizzle-fill LDS with B (coalesced global reads)
  for (int idx = tid; idx < K * 128; idx += 256) {
    int kr = idx >> 7;          // row of B
    int col = idx & 127;        // col of B
    int ks = kr >> 2;           // k-step
    int c = col >> 4;           // col tile
    int lane = (col & 15) | ((kr & 2) << 3);
    int sub = kr & 1;
    bs[((ks * 32 + lane) * 8 + c) * 2 + sub] = Bmat[idx];
  }
  __syncthreads();

  const int wave = tid >> 5;
  const int lane = tid & 31;
  const int row0 = (blockIdx.x * 8 + wave) * 16;
  if (row0 >= M) return;        // wave-uniform; EXEC stays all-1 for WMMA

  const int m  = lane & 15;
  const int kk = (lane >> 4) << 1;  // 0 or 2

  v8f acc[8] = {};

  const float* arow = Amat + (size_t)(row0 + m) * K + kk;  // 8B-aligned steps
  const float* blane = bs + lane * 16;                     // 64B-aligned

#pragma unroll 4
  for (int ks = 0; ks < K / 4; ++ks) {
    v2f a = *(const v2f*)(arow + ks * 4);                  // global_load_b64
    const float* bp = blane + ks * 512;
    v4f q0 = *(const v4f*)(bp);                            // ds_load_b128 x4
    v4f q1 = *(const v4f*)(bp + 4);
    v4f q2 = *(const v4f*)(bp + 8);
    v4f q3 = *(const v4f*)(bp + 12);
    acc[0] = __builtin_amdgcn_wmma_f32_16x16x4_f32(false, a, false, q0.xy, (short)0, acc[0], false, false);
    acc[1] = __builtin_amdgcn_wmma_f32_16x16x4_f32(false, a, false, q0.zw, (short)0, acc[1], false, false);
    acc[2] = __builtin_amdgcn_wmma_f32_16x16x4_f32(false, a, false, q1.xy, (short)0, acc[2], false, false);
    acc[3] = __builtin_amdgcn_wmma_f32_16x16x4_f32(false, a, false, q1.zw, (short)0, acc[3], false, false);
    acc[4] = __builtin_amdgcn_wmma_f32_16x16x4_f32(false, a, false, q2.xy, (short)0, acc[4], false, false);
    acc[5] = __builtin_amdgcn_wmma_f32_16x16x4_f32(false, a, false, q2.zw, (short)0, acc[5], false, false);
    acc[6] = __builtin_amdgcn_wmma_f32_16x16x4_f32(false, a, false, q3.xy, (short)0, acc[6], false, false);
    acc[7] = __builtin_amdgcn_wmma_f32_16x16x4_f32(false, a, false, q3.zw, (short)0, acc[7], false, false);
  }

  const int mo = (lane >> 4) * 8;
#pragma unroll
  for (int c = 0; c < 8; ++c) {
#pragma unroll
    for (int r = 0; r < 8; ++r) {
      Cmat[(size_t)(row0 + mo + r) * 128 + c * 16 + m] = acc[c][r];
    }
  }
}

// ---------------------------------------------------------------------------
// el[n,h] = sum_f h[n,h,f]*attn_l[h,f] ; er likewise. One thread per (n,h).
// ---------------------------------------------------------------------------
__global__ void gat_attn_proj(const float* __restrict__ h,
                              const float* __restrict__ al,
                              const float* __restrict__ ar,
                              float* __restrict__ el, float* __restrict__ er,
                              int NH) {
  int i = blockIdx.x * blockDim.x + threadIdx.x;
  if (i >= NH) return;
  int n = i >> 2, hh = i & 3;
  const float* hp  = h + (size_t)n * 128 + hh * 32;
  const float* alp = al + hh * 32;
  const float* arp = ar + hh * 32;
  float sl = 0.f, sr = 0.f;
#pragma unroll
  for (int f = 0; f < 32; ++f) {
    float v = hp[f];
    sl += v * alp[f];
    sr += v * arp[f];
  }
  el[i] = sl;
  er[i] = sr;
}

// ---- edge pass A: segment max over incoming edges -------------------------
__global__ void gat_edge_max(const int* __restrict__ src, const int* __restrict__ dst,
                             const float* __restrict__ el, const float* __restrict__ er,
                             unsigned* __restrict__ emax, int E4) {
  int i = blockIdx.x * blockDim.x + threadIdx.x;
  if (i >= E4) return;
  int e = i >> 2, hh = i & 3;
  int s = src[e], d = dst[e];
  float ev = lrelu(el[s * 4 + hh] + er[d * 4 + hh]);
  atomicMax(&emax[d * 4 + hh], fenc(ev));
}

// ---- edge pass B: denom = segment sum of exp(e - max) ---------------------
__global__ void gat_edge_denom(const int* __restrict__ src, const int* __restrict__ dst,
                               const float* __restrict__ el, const float* __restrict__ er,
                               const unsigned* __restrict__ emax,
                               float* __restrict__ denom, int E4) {
  int i = blockIdx.x * blockDim.x + threadIdx.x;
  if (i >= E4) return;
  int e = i >> 2, hh = i & 3;
  int s = src[e], d = dst[e];
  float ev = lrelu(el[s * 4 + hh] + er[d * 4 + hh]);
  float mx = fdec(emax[d * 4 + hh]);
  atomicAdd(&denom[d * 4 + hh], expf(ev - mx));
}

// ---- edge pass C: rst[dst] += alpha * h[src]; 128 threads per edge --------
__global__ void gat_edge_scatter(const int* __restrict__ src, const int* __restrict__ dst,
                                 const float* __restrict__ el, const float* __restrict__ er,
                                 const unsigned* __restrict__ emax,
                                 const float* __restrict__ denom,
                                 const float* __restrict__ h,
                                 float* __restrict__ rst, int E) {
  int gid = blockIdx.x * 256 + threadIdx.x;
  int e = gid >> 7;
  if (e >= E) return;
  int t = gid & 127;
  int hh = t >> 5;
  int s = src[e], d = dst[e];
  float ev = lrelu(el[s * 4 + hh] + er[d * 4 + hh]);
  float mx = fdec(emax[d * 4 + hh]);
  float ex = expf(ev - mx);
  float alpha = ex / fmaxf(denom[d * 4 + hh], 1e-9f);
  atomicAdd(&rst[(size_t)d * 128 + t], h[(size_t)s * 128 + t] * alpha);
}

// ---- layer-1 epilogue: head mean + bias, LayerNorm, ReLU (wave per node) --
__global__ void gat_finalize1(const float* __restrict__ rst,
                              const float* __restrict__ bias,
                              const float* __restrict__ lns,
                              const float* __restrict__ lnb,
                              float* __restrict__ hmid, int Nn) {
  int n = blockIdx.x * 8 + (threadIdx.x >> 5);
  int f = threadIdx.x & 31;
  if (n >= Nn) return;
  float x = 0.f;
#pragma unroll
  for (int hh = 0; hh < 4; ++hh)
    x += rst[(size_t)n * 128 + hh * 32 + f] + bias[hh * 32 + f];
  x *= 0.25f;
  // wave32 reductions (warpSize == 32 on gfx1250)
  float mu = x;
#pragma unroll
  for (int m = 16; m >= 1; m >>= 1) mu += __shfl_xor(mu, m, 32);
  mu *= (1.f / 32.f);
  float dx = x - mu;
  float var = dx * dx;
#pragma unroll
  for (int m = 16; m >= 1; m >>= 1) var += __shfl_xor(var, m, 32);
  var *= (1.f / 32.f);
  float y = dx * rsqrtf(var + LN_EPS) * lns[f] + lnb[f];
  hmid[(size_t)n * 32 + f] = fmaxf(y, 0.f);
}

// ---- layer-2 epilogue: head mean + bias ----------------------------------
__global__ void gat_finalize2(const float* __restrict__ rst,
                              const float* __restrict__ bias,
                              float* __restrict__ out, int NF) {
  int i = blockIdx.x * blockDim.x + threadIdx.x;
  if (i >= NF) return;
  int n = i >> 5, f = i & 31;
  float x = 0.f;
#pragma unroll
  for (int hh = 0; hh < 4; ++hh)
    x += rst[(size_t)n * 128 + hh * 32 + f] + bias[hh * 32 + f];
  out[i] = x * 0.25f;
}

// ---- flop counter scalar (reference returns a tuple (h, flops)) -----------
__global__ void gat_flops(const int* __restrict__ in_deg, int Nn, float* out) {
  __shared__ float ssum[256];
  float s = 0.f;
  for (int i = threadIdx.x; i < Nn; i += 256) s += (float)in_deg[i];
  ssum[threadIdx.x] = s;
  __syncthreads();
  for (int m = 128; m > 0; m >>= 1) {
    if ((int)threadIdx.x < m) ssum[threadIdx.x] += ssum[threadIdx.x + m];
    __syncthreads();
  }
  if (threadIdx.x == 0) {
    double ne = (double)ssum[0];
    double fl = (4.0 * ne * (6.0 * 128 * 32 + 6 * 32 + 2) +
                 4.0 * ne * (6.0 * 32 * 32 + 6 * 32 + 2)) / 1e12;
    out[0] = (float)fl;
  }
}

// ---------------------------------------------------------------------------
extern "C" void kernel_launch(void* const* d_in, const int* in_sizes, int n_in,
                              void* d_out, int out_size, void* d_ws, size_t ws_size,
                              hipStream_t stream) {
  const float* feat = (const float*)d_in[0];
  const float* W1   = (const float*)d_in[1];
  const float* al1  = (const float*)d_in[2];
  const float* ar1  = (const float*)d_in[3];
  const float* b1   = (const float*)d_in[4];
  const float* lns  = (const float*)d_in[5];
  const float* lnb  = (const float*)d_in[6];
  const float* W2   = (const float*)d_in[7];
  const float* al2  = (const float*)d_in[8];
  const float* ar2  = (const float*)d_in[9];
  const float* b2   = (const float*)d_in[10];
  const int*   src  = (const int*)d_in[11];
  const int*   dst  = (const int*)d_in[12];
  const int*   indeg= (const int*)d_in[13];

  const int Nn = in_sizes[13];          // 50000
  const int E  = in_sizes[11];          // 800000
  const int NH = Nn * 4;

  // workspace layout (buffers reused across layers; total ~58 MB)
  char* p = (char*)d_ws;
  float*    hbuf  = (float*)p;   p += (size_t)Nn * 128 * sizeof(float); // h1 / h2
  float*    rst   = (float*)p;   p += (size_t)Nn * 128 * sizeof(float); // rst1 / rst2
  float*    hmid  = (float*)p;   p += (size_t)Nn * 32  * sizeof(float);
  float*    el    = (float*)p;   p += (size_t)NH * sizeof(float);
  float*    er    = (float*)p;   p += (size_t)NH * sizeof(float);
  unsigned* emax  = (unsigned*)p;p += (size_t)NH * sizeof(unsigned);
  float*    denom = (float*)p;   p += (size_t)NH * sizeof(float);
  (void)ws_size; (void)n_in; (void)out_size;

  float* outh = (float*)d_out;

  dim3 b256(256);
  const int rowTiles   = (Nn + 15) / 16;       // 3125
  const int gemmBlocks = (rowTiles + 7) / 8;   // 8 waves/block, 1 tile/wave
  const int e4Blocks   = (E * 4 + 255) / 256;
  const int scBlocks   = (E * 128 + 255) / 256;

  // ---------------- layer 1 ----------------
  hipMemsetAsync(rst,   0, (size_t)Nn * 128 * sizeof(float), stream);
  hipMemsetAsync(emax,  0, (size_t)NH * sizeof(unsigned), stream);   // enc(-inf) > 0, 0 is identity
  hipMemsetAsync(denom, 0, (size_t)NH * sizeof(float), stream);

  gat_gemm_wmma<128><<<gemmBlocks, b256, 0, stream>>>(feat, W1, hbuf, Nn);
  gat_attn_proj<<<(NH + 255) / 256, b256, 0, stream>>>(hbuf, al1, ar1, el, er, NH);
  gat_edge_max<<<e4Blocks, b256, 0, stream>>>(src, dst, el, er, emax, E * 4);
  gat_edge_denom<<<e4Blocks, b256, 0, stream>>>(src, dst, el, er, emax, denom, E * 4);
  gat_edge_scatter<<<scBlocks, b256, 0, stream>>>(src, dst, el, er, emax, denom, hbuf, rst, E);
  gat_finalize1<<<(Nn + 7) / 8, b256, 0, stream>>>(rst, b1, lns, lnb, hmid, Nn);

  // ---------------- layer 2 ----------------
  hipMemsetAsync(rst,   0, (size_t)Nn * 128 * sizeof(float), stream);
  hipMemsetAsync(emax,  0, (size_t)NH * sizeof(unsigned), stream);
  hipMemsetAsync(denom, 0, (size_t)NH * sizeof(float), stream);

  gat_gemm_wmma<32><<<gemmBlocks, b256, 0, stream>>>(hmid, W2, hbuf, Nn);
  gat_attn_proj<<<(NH + 255) / 256, b256, 0, stream>>>(hbuf, al2, ar2, el, er, NH);
  gat_edge_max<<<e4Blocks, b256, 0, stream>>>(src, dst, el, er, emax, E * 4);
  gat_edge_denom<<<e4Blocks, b256, 0, stream>>>(src, dst, el, er, emax, denom, E * 4);
  gat_edge_scatter<<<scBlocks, b256, 0, stream>>>(src, dst, el, er, emax, denom, hbuf, rst, E);
  gat_finalize2<<<(Nn * 32 + 255) / 256, b256, 0, stream>>>(rst, b2, outh, Nn * 32);

  gat_flops<<<1, 256, 0, stream>>>(indeg, Nn, outh + (size_t)Nn * 32);
}